// DetectionSampler_84628035601038
// MI455X (gfx1250) — compile-verified
//
#include <hip/hip_runtime.h>
#include <hip/hip_bf16.h>

// Problem constants (match reference)
#define BN    4
#define DCH   128
#define HH    512
#define WW    512
#define HWP   (HH * WW)          // 262144
#define CELL  16
#define BORD  16
#define HC    30
#define WC    30
#define NPB   (HC * WC)          // 900 keypoints per batch
#define NKP   (BN * NPB)         // 3600 keypoints total
#define NCOL  (NKP + 1)          // 3601 score columns
#define NT    (NKP / 16)         // 225 WMMA tiles per dim
#define JTPW  3                  // j-tiles per wave (225 = 75*3, no remainder)
#define NJW   (NT / JTPW)        // 75 wave-columns
#define KOFF  13                 // |{(i,j): i^2+j^2<=4}|

typedef __attribute__((ext_vector_type(2))) float v2f;
typedef __attribute__((ext_vector_type(8))) float v8f;

// Offsets in the exact enumeration order of the reference:
// (i,j) for i in -2..2, j in -2..2 with i*i+j*j<=4 ; i adds to x, j adds to y.
__constant__ int c_offx[KOFF] = {-2,-1,-1,-1, 0, 0, 0, 0, 0, 1, 1, 1, 2};
__constant__ int c_offy[KOFF] = { 0,-1, 0, 1,-2,-1, 0, 1, 2,-1, 0, 1, 0};

// ---------------------------------------------------------------------------
// Kernel 1: per-cell argmax over both detection maps (1 thread per cell).
// First-occurrence tie-break (strict >) matches jnp.argmax row-major order.
// ---------------------------------------------------------------------------
__global__ __launch_bounds__(256) void sample_cells_kernel(
    const float* __restrict__ det1, const float* __restrict__ det2,
    int* __restrict__ kp1y, int* __restrict__ kp1x,
    int* __restrict__ kp2y, int* __restrict__ kp2x)
{
    int t = blockIdx.x * blockDim.x + threadIdx.x;
    if (t >= 2 * NKP) return;
    const bool first = (t < NKP);
    const float* det = first ? det1 : det2;
    int* ky = first ? kp1y : kp2y;
    int* kx = first ? kp1x : kp2x;
    int p  = first ? t : (t - NKP);
    int b  = p / NPB;
    int r  = p % NPB;
    int hc = r / WC;
    int wc = r % WC;
    const float* base = det + (size_t)b * HWP
                            + (size_t)(BORD + hc * CELL) * WW
                            + (BORD + wc * CELL);
    float best = -3.4e38f;
    int   bi   = 0;
    for (int idx = 0; idx < CELL * CELL; ++idx) {
        float v = base[(idx >> 4) * WW + (idx & 15)];
        if (v > best) { best = v; bi = idx; }
    }
    ky[p] = BORD + hc * CELL + (bi >> 4);
    kx[p] = BORD + wc * CELL + (bi & 15);
}

// ---------------------------------------------------------------------------
// Kernel 2: one wave (32 lanes) per det1 keypoint.
//  - gather s_des1 (4 channels per lane) into ws
//  - xy2 / in-bounds mask
//  - 13-offset neighborhood dot products (wave reduction via __shfl_xor)
//  - pscores -> scores[:,0], qlt, mask outputs
// ---------------------------------------------------------------------------
__global__ __launch_bounds__(256) void prep_kp1_kernel(
    const float* __restrict__ des1, const float* __restrict__ des2,
    const float* __restrict__ qlt1, const float* __restrict__ qlt2,
    const float* __restrict__ aflow,
    const int* __restrict__ kp1y, const int* __restrict__ kp1x,
    int* __restrict__ xy2x, int* __restrict__ xy2y,
    float* __restrict__ sdes1,
    float* __restrict__ out_scores, float* __restrict__ out_mask,
    float* __restrict__ out_qlt)
{
    int gtid = blockIdx.x * blockDim.x + threadIdx.x;
    int wave = gtid >> 5;
    int lane = gtid & 31;
    if (wave >= NKP) return;
    int p = wave;
    int b = p / NPB;
    int y1 = kp1y[p];
    int x1 = kp1x[p];

    // gather 128-d descriptor: channel c = lane + 32*t
    size_t d1base = (size_t)b * DCH * HWP + (size_t)y1 * WW + x1;
    float s[4];
#pragma unroll
    for (int t = 0; t < 4; ++t) {
        int c = lane + 32 * t;
        s[t] = des1[d1base + (size_t)c * HWP];
        sdes1[(size_t)p * DCH + c] = s[t];
    }

    // flow target (trunc-toward-zero int cast matches astype(int32))
    size_t apix = (size_t)b * 2 * HWP + (size_t)y1 * WW + x1;
    float fx = aflow[apix];
    float fy = aflow[apix + HWP];
    int x2 = (int)(fx + 0.5f);
    int y2 = (int)(fy + 0.5f);

    // 13-offset dot products, tracking first-occurrence argmax
    size_t d2base = (size_t)b * DCH * HWP;
    float best = -3.4e38f;
    int bestk = 0;
    for (int k = 0; k < KOFF; ++k) {
        int xp = min(max(x2 + c_offx[k], 0), WW - 1);
        int yp = min(max(y2 + c_offy[k], 0), HH - 1);
        size_t q = d2base + (size_t)yp * WW + xp;
        float part = 0.0f;
#pragma unroll
        for (int t = 0; t < 4; ++t) {
            int c = lane + 32 * t;
            part += s[t] * des2[q + (size_t)c * HWP];
        }
#pragma unroll
        for (int off = 16; off > 0; off >>= 1)
            part += __shfl_xor(part, off, 32);
        if (part > best) { best = part; bestk = k; }   // uniform across wave
    }

    if (lane == 0) {
        xy2x[p] = x2;
        xy2y[p] = y2;
        out_mask[p] = (x2 >= 0 && y2 >= 0 && x2 < WW && y2 < HH) ? 1.0f : 0.0f;
        int xs = min(max(x2 + c_offx[bestk], 0), WW - 1);
        int ys = min(max(y2 + c_offy[bestk], 0), HH - 1);
        float qv = 0.5f * (qlt1[(size_t)b * HWP + (size_t)y1 * WW + x1] +
                           qlt2[(size_t)b * HWP + (size_t)ys * WW + xs]);
        out_qlt[p] = qv;
        out_scores[(size_t)p * NCOL] = best;   // pscores column
    }
}

// ---------------------------------------------------------------------------
// Kernel 3: one wave per det2 keypoint -> gather distractor descriptors.
// ---------------------------------------------------------------------------
__global__ __launch_bounds__(256) void prep_kp2_kernel(
    const float* __restrict__ des2,
    const int* __restrict__ kp2y, const int* __restrict__ kp2x,
    float* __restrict__ distr)
{
    int gtid = blockIdx.x * blockDim.x + threadIdx.x;
    int wave = gtid >> 5;
    int lane = gtid & 31;
    if (wave >= NKP) return;
    int p = wave;
    int b = p / NPB;
    int yd = kp2y[p];
    int xd = kp2x[p];
    size_t base = (size_t)b * DCH * HWP + (size_t)yd * WW + xd;
#pragma unroll
    for (int t = 0; t < 4; ++t) {
        int c = lane + 32 * t;
        distr[(size_t)p * DCH + c] = des2[base + (size_t)c * HWP];
    }
}

// ---------------------------------------------------------------------------
// Kernel 4: 3600x3600x128 fp32 GEMM via V_WMMA_F32_16X16X4_F32 + distance
// masking -> scores[:, 1:]. One 16x48 tile per wave: the A fragment is
// reused (in registers, reuse hints legally 0) across 3 accumulators, so
// 4 fragment loads feed 3 WMMAs. K=128 chained in 32 steps per tile.
// Fragment layout (32-bit A 16x4): lanes 0-15 hold K={0,1}, lanes 16-31
// hold K={2,3} for row M = lane&15 -> aligned float2 loads. Tile guard is
// wave-uniform so EXEC is all-ones at every WMMA. Output stores are
// non-temporal: written once, never re-read -> stream to HBM.
// ---------------------------------------------------------------------------
__global__ __launch_bounds__(256) void gemm_mask_kernel(
    const float* __restrict__ sdes1, const float* __restrict__ distr,
    const int* __restrict__ xy2x, const int* __restrict__ xy2y,
    const int* __restrict__ kp2x, const int* __restrict__ kp2y,
    float* __restrict__ out_scores)
{
    int gtid = blockIdx.x * blockDim.x + threadIdx.x;
    int wave = gtid >> 5;
    if (wave >= NT * NJW) return;          // wave-uniform exit
    int lane = threadIdx.x & 31;
    int it = wave / NJW;
    int jw = wave % NJW;
    int i0 = it * 16;
    int j0 = jw * (JTPW * 16);
    int half = lane >> 4;                  // 0: K={0,1}, 1: K={2,3}
    int m    = lane & 15;                  // row/col within tile

    const float* arow = sdes1 + (size_t)(i0 + m) * DCH + 2 * half;
    const float* brow = distr + (size_t)(j0 + m) * DCH + 2 * half;

    // touch the back halves of the streamed panels (global_prefetch_b8)
    __builtin_prefetch(arow + 64, 0, 1);
    __builtin_prefetch(brow + 64, 0, 1);
    __builtin_prefetch(brow + 16 * DCH + 64, 0, 1);
    __builtin_prefetch(brow + 32 * DCH + 64, 0, 1);

    v8f acc0 = {}, acc1 = {}, acc2 = {};
#pragma unroll 4
    for (int kk = 0; kk < DCH; kk += 4) {
        v2f a  = *(const v2f*)(arow + kk);                 // 8B aligned
        v2f b0 = *(const v2f*)(brow + kk);
        v2f b1 = *(const v2f*)(brow + 16 * DCH + kk);
        v2f b2 = *(const v2f*)(brow + 32 * DCH + kk);
        acc0 = __builtin_amdgcn_wmma_f32_16x16x4_f32(
            false, a, false, b0, (short)0, acc0, false, false);
        acc1 = __builtin_amdgcn_wmma_f32_16x16x4_f32(
            false, a, false, b1, (short)0, acc1, false, false);
        acc2 = __builtin_amdgcn_wmma_f32_16x16x4_f32(
            false, a, false, b2, (short)0, acc2, false, false);
    }

    // Row metadata shared by all three j-tiles.
    // C/D layout: VGPR r -> M = r (lanes 0-15) or r+8 (lanes 16-31), N = lane&15
    int rx[8], ry[8], rb[8];
#pragma unroll
    for (int r = 0; r < 8; ++r) {
        int row = i0 + r + 8 * half;
        rx[r] = xy2x[row];
        ry[r] = xy2y[row];
        rb[r] = row / NPB;
    }

#pragma unroll
    for (int jsub = 0; jsub < JTPW; ++jsub) {
        int col = j0 + 16 * jsub + m;
        int xdv = kp2x[col];
        int ydv = kp2y[col];
        int bd  = col / NPB;
        const v8f& acc = (jsub == 0) ? acc0 : (jsub == 1) ? acc1 : acc2;
#pragma unroll
        for (int r = 0; r < 8; ++r) {
            int row = i0 + r + 8 * half;
            int dx  = xdv - rx[r];
            int dy  = ydv - ry[r];
            int dis2 = dx * dx + dy * dy + ((bd != rb[r]) ? 4 : 0);
            float v = (dis2 < 4) ? 0.0f : acc[r];
            __builtin_nontemporal_store(v, &out_scores[(size_t)row * NCOL + 1 + col]);
        }
    }
}

// ---------------------------------------------------------------------------
// Kernel 5: one-hot labels fill (write-once stream -> non-temporal).
// ---------------------------------------------------------------------------
__global__ __launch_bounds__(256) void fill_labels_kernel(float* __restrict__ labels)
{
    size_t total = (size_t)NKP * NCOL;
    size_t idx = (size_t)blockIdx.x * blockDim.x + threadIdx.x;
    if (idx >= total) return;
    float v = ((idx % NCOL) == 0) ? 1.0f : 0.0f;
    __builtin_nontemporal_store(v, &labels[idx]);
}

// ---------------------------------------------------------------------------
extern "C" void kernel_launch(void* const* d_in, const int* in_sizes, int n_in,
                              void* d_out, int out_size, void* d_ws, size_t ws_size,
                              hipStream_t stream) {
    const float* des1  = (const float*)d_in[0];
    const float* det1  = (const float*)d_in[1];
    const float* qlt1  = (const float*)d_in[2];
    const float* des2  = (const float*)d_in[3];
    const float* det2  = (const float*)d_in[4];
    const float* qlt2  = (const float*)d_in[5];
    const float* aflow = (const float*)d_in[6];

    // workspace layout
    int*   kp1y  = (int*)d_ws;
    int*   kp1x  = kp1y + NKP;
    int*   kp2y  = kp1x + NKP;
    int*   kp2x  = kp2y + NKP;
    int*   xy2x  = kp2x + NKP;
    int*   xy2y  = xy2x + NKP;
    float* sdes1 = (float*)(xy2y + NKP);            // NKP*128
    float* distr = sdes1 + (size_t)NKP * DCH;       // NKP*128

    // output layout: scores | labels | mask | qlt  (all fp32)
    float* scores = (float*)d_out;
    float* labels = scores + (size_t)NKP * NCOL;
    float* maskp  = labels + (size_t)NKP * NCOL;
    float* qltp   = maskp + NKP;

    // 1) cell argmax for both maps
    {
        int threads = 2 * NKP;
        sample_cells_kernel<<<(threads + 255) / 256, 256, 0, stream>>>(
            det1, det2, kp1y, kp1x, kp2y, kp2x);
    }
    // 2) det1 keypoints: descriptors, xy2, mask, pscores, qlt
    {
        int threads = NKP * 32;
        prep_kp1_kernel<<<(threads + 255) / 256, 256, 0, stream>>>(
            des1, des2, qlt1, qlt2, aflow, kp1y, kp1x,
            xy2x, xy2y, sdes1, scores, maskp, qltp);
    }
    // 3) det2 keypoints: distractor descriptors
    {
        int threads = NKP * 32;
        prep_kp2_kernel<<<(threads + 255) / 256, 256, 0, stream>>>(
            des2, kp2y, kp2x, distr);
    }
    // 4) WMMA GEMM (16x48 tiles, A-reuse) + distance masking -> scores[:, 1:]
    {
        long long threads = (long long)NT * NJW * 32;
        gemm_mask_kernel<<<(int)((threads + 255) / 256), 256, 0, stream>>>(
            sdes1, distr, xy2x, xy2y, kp2x, kp2y, scores);
    }
    // 5) labels
    {
        size_t total = (size_t)NKP * NCOL;
        fill_labels_kernel<<<(int)((total + 255) / 256), 256, 0, stream>>>(labels);
    }
}